// GravNet_90194313216185
// MI455X (gfx1250) — compile-verified
//
#include <hip/hip_runtime.h>

// ---------------- problem constants ----------------
constexpr int B_    = 8;
constexpr int N_    = 4096;
constexpr int F_IN  = 32;
constexpr int K_    = 40;
constexpr int S_    = 4;
constexpr int P_    = 22;
constexpr int F_OUT = 48;
constexpr int FSTR  = 24;          // feats row stride (22 padded to 24 for float4 I/O)
constexpr int UDIM  = F_IN + 2*P_; // 76
constexpr int GSTR  = 20;          // d2 stage row stride: 16B-aligned rows, conflict-free

typedef float v2f __attribute__((ext_vector_type(2)));
typedef float v8f __attribute__((ext_vector_type(8)));

// ---------------- kernel A: projections ----------------
// feats = x@Wf + bf ; coords = x@Ws + bs ; sq = |coords|^2
__global__ __launch_bounds__(256) void gravnet_proj(
    const float* __restrict__ x,  const float* __restrict__ Wf, const float* __restrict__ bf,
    const float* __restrict__ Ws, const float* __restrict__ bs,
    float* __restrict__ coords, float* __restrict__ sqv, float* __restrict__ feats)
{
    int gid = blockIdx.x * 256 + threadIdx.x;     // b*N + n
    if (gid >= B_ * N_) return;

    float xv[F_IN];
    const float4* xp = (const float4*)(x + (size_t)gid * F_IN);
#pragma unroll
    for (int i = 0; i < F_IN / 4; ++i) {
        float4 v = xp[i];
        xv[4*i+0] = v.x; xv[4*i+1] = v.y; xv[4*i+2] = v.z; xv[4*i+3] = v.w;
    }

    // coords (S=4) + squared norm
    float c[S_];
#pragma unroll
    for (int s = 0; s < S_; ++s) c[s] = bs[s];
#pragma unroll
    for (int i = 0; i < F_IN; ++i) {
#pragma unroll
        for (int s = 0; s < S_; ++s) c[s] = fmaf(xv[i], Ws[i*S_ + s], c[s]);
    }
    ((float4*)coords)[gid] = make_float4(c[0], c[1], c[2], c[3]);
    sqv[gid] = c[0]*c[0] + c[1]*c[1] + c[2]*c[2] + c[3]*c[3];

    // feats (P=22, padded to 24)
    float f[FSTR];
#pragma unroll
    for (int p = 0; p < P_; ++p) {
        float a = bf[p];
#pragma unroll
        for (int i = 0; i < F_IN; ++i) a = fmaf(xv[i], Wf[i*P_ + p], a);
        f[p] = a;
    }
    f[22] = 0.f; f[23] = 0.f;
    float4* fo = (float4*)(feats + (size_t)gid * FSTR);
#pragma unroll
    for (int q = 0; q < FSTR / 4; ++q)
        fo[q] = make_float4(f[4*q+0], f[4*q+1], f[4*q+2], f[4*q+3]);
}

// ---------------- kernel B: WMMA distance tiles -> top-K -> aggregate -> out ----------------
// Block: 64 threads (2 waves). Wave w owns 32 rows; each thread owns one row's top-K.
// Gram tile via V_WMMA_F32_16X16X4_F32 (S=4 == WMMA K, fp32 exact). A is pre-scaled by -2
// so D = -2*<c_n,c_m> and d2 = |n|^2 + |m|^2 + D with plain adds.
__global__ __launch_bounds__(64) void gravnet_main(
    const float* __restrict__ x,  const float* __restrict__ Wo, const float* __restrict__ bo,
    const float* __restrict__ coords, const float* __restrict__ sqv,
    const float* __restrict__ feats, float* __restrict__ out)
{
    __shared__ float sWo[UDIM * F_OUT];    // 76*48
    __shared__ float sBo[F_OUT];
    __shared__ float gstage[2][32 * GSTR]; // per-wave private stage; wave-local sync only
    __shared__ float skd[64 * K_];         // per-thread top-K dump (dynamic indexing)
    __shared__ int   ski[64 * K_];

    const int b    = blockIdx.y;
    const int wave = threadIdx.x >> 5;
    const int lane = threadIdx.x & 31;
    const int col  = lane & 15;
    const int hf   = lane >> 4;                       // 0: K={0,1} ; 1: K={2,3}
    const int rowBase = blockIdx.x * 64 + wave * 32;  // this wave's first row
    const int grow    = blockIdx.x * 64 + (int)threadIdx.x; // this thread's row n

    for (int i = threadIdx.x; i < UDIM * F_OUT; i += 64) sWo[i] = Wo[i];
    if (threadIdx.x < F_OUT) sBo[threadIdx.x] = bo[threadIdx.x];

    const float* cb  = coords + (size_t)b * N_ * 4;
    const float* sqb = sqv    + (size_t)b * N_;

    // A tiles: 16x4 f32, lane = M (0..15), half selects K pair {0,1}/{2,3}; pre-scale by -2
    v2f a0 = *(const v2f*)(cb + (size_t)(rowBase      + col) * 4 + hf * 2);
    v2f a1 = *(const v2f*)(cb + (size_t)(rowBase + 16 + col) * 4 + hf * 2);
    a0 *= -2.0f;
    a1 *= -2.0f;

    // per-lane row norms matching D-matrix layout (vgpr r -> row r or r+8)
    float srA[8], srB[8];
#pragma unroll
    for (int r = 0; r < 8; ++r) {
        srA[r] = sqb[rowBase      + (hf ? 8 : 0) + r];
        srB[r] = sqb[rowBase + 16 + (hf ? 8 : 0) + r];
    }

    // register-resident top-K (replace-worst; all accesses constant-indexed)
    float kd[K_]; int ki[K_];
#pragma unroll
    for (int j = 0; j < K_; ++j) { kd[j] = 3.0e38f; ki[j] = -1; }
    float worst = 3.0e38f; int wslot = 0;

    float* gst = &gstage[wave][0];

    for (int t = 0; t < N_ / 16; ++t) {
        const int cb0 = t * 16;
        // B tile: 4x16 f32, lane = N column, half selects K pair
        v2f bm   = *(const v2f*)(cb + (size_t)(cb0 + col) * 4 + hf * 2);
        float sqc = sqb[cb0 + col];

        v8f d0 = {0.f,0.f,0.f,0.f,0.f,0.f,0.f,0.f};
        v8f d1 = {0.f,0.f,0.f,0.f,0.f,0.f,0.f,0.f};
        d0 = __builtin_amdgcn_wmma_f32_16x16x4_f32(false, a0, false, bm, (short)0, d0, false, false);
        d1 = __builtin_amdgcn_wmma_f32_16x16x4_f32(false, a1, false, bm, (short)0, d1, false, false);

        // stage full d2 = |n|^2 + |m|^2 - 2<c_n,c_m>   (wave-private region)
#pragma unroll
        for (int r = 0; r < 8; ++r) {
            int row0 = hf ? (r + 8) : r;
            gst[row0 * GSTR + col]        = (srA[r] + sqc) + d0[r];
            gst[(row0 + 16) * GSTR + col] = (srB[r] + sqc) + d1[r];
        }
        // wave-local fence: DS is in-order per wave; no inter-wave barrier needed
        asm volatile("s_wait_dscnt 0x0" ::: "memory");

        // thread 'lane' owns staged row 'lane': pull 4 candidates per ds_load_b128
#pragma unroll 1
        for (int g = 0; g < 4; ++g) {
            float4 q = *(const float4*)&gst[lane * GSTR + g * 4];
            int cbase = cb0 + g * 4;
            float cv[4] = { q.x, q.y, q.z, q.w };
#pragma unroll
            for (int c = 0; c < 4; ++c) {
                float dv = cv[c];
                int   ci = cbase + c;
                if (dv < worst) {
#pragma unroll
                    for (int j = 0; j < K_; ++j) if (j == wslot) { kd[j] = dv; ki[j] = ci; }
                    worst = kd[0]; wslot = 0;
#pragma unroll
                    for (int j = 1; j < K_; ++j) if (kd[j] > worst) { worst = kd[j]; wslot = j; }
                }
            }
        }
    }

    // dump list to LDS so the gather loop can index it dynamically (no spills)
#pragma unroll
    for (int j = 0; j < K_; ++j) {
        skd[threadIdx.x * K_ + j] = kd[j];
        ski[threadIdx.x * K_ + j] = ki[j];
    }

    // aggregation: max & mean of w * feats over 39 neighbors (order invariant; drop self)
    float maxv[P_], sumv[P_];
#pragma unroll
    for (int p = 0; p < P_; ++p) { maxv[p] = -3.0e38f; sumv[p] = 0.f; }

    const float* fb = feats + (size_t)b * N_ * FSTR;
#pragma unroll 1
    for (int j = 0; j < K_; ++j) {
        int ni = ski[threadIdx.x * K_ + j];
        if (ni < 0 || ni == grow) continue;  // exclude self (the [:, :, 1:] drop)
        float w = __expf(-10.0f * fabsf(skd[threadIdx.x * K_ + j]));
        const float4* fp = (const float4*)(fb + (size_t)ni * FSTR);
        float4 f0 = fp[0], f1 = fp[1], f2 = fp[2], f3 = fp[3], f4 = fp[4], f5 = fp[5];
        float fv[FSTR] = { f0.x,f0.y,f0.z,f0.w, f1.x,f1.y,f1.z,f1.w,
                           f2.x,f2.y,f2.z,f2.w, f3.x,f3.y,f3.z,f3.w,
                           f4.x,f4.y,f4.z,f4.w, f5.x,f5.y,f5.z,f5.w };
#pragma unroll
        for (int p = 0; p < P_; ++p) {
            float v = fv[p] * w;
            maxv[p] = fmaxf(maxv[p], v);
            sumv[p] += v;
        }
    }

    // updated = [x, max, mean]; out = tanh(updated @ Wo + bo)
    float u[UDIM];
    const float4* xp = (const float4*)(x + ((size_t)b * N_ + grow) * F_IN);
#pragma unroll
    for (int i = 0; i < F_IN / 4; ++i) {
        float4 v = xp[i];
        u[4*i+0] = v.x; u[4*i+1] = v.y; u[4*i+2] = v.z; u[4*i+3] = v.w;
    }
#pragma unroll
    for (int p = 0; p < P_; ++p) {
        u[F_IN + p]      = maxv[p];
        u[F_IN + P_ + p] = sumv[p] * (1.0f / 39.0f);
    }

    __syncthreads();  // sWo/sBo preload visible to all threads (only barrier in the kernel)

    float* orow = out + ((size_t)b * N_ + grow) * F_OUT;
#pragma unroll 1
    for (int f = 0; f < F_OUT; ++f) {
        float acc = sBo[f];
#pragma unroll
        for (int j = 0; j < UDIM; ++j) acc = fmaf(u[j], sWo[j * F_OUT + f], acc); // uniform LDS -> broadcast
        orow[f] = tanhf(acc);
    }
}

// ---------------- launch ----------------
extern "C" void kernel_launch(void* const* d_in, const int* in_sizes, int n_in,
                              void* d_out, int out_size, void* d_ws, size_t ws_size,
                              hipStream_t stream) {
    const float* x  = (const float*)d_in[0];
    const float* Wf = (const float*)d_in[1];
    const float* bf = (const float*)d_in[2];
    const float* Ws = (const float*)d_in[3];
    const float* bs = (const float*)d_in[4];
    const float* Wo = (const float*)d_in[5];
    const float* bo = (const float*)d_in[6];
    float* outp = (float*)d_out;

    // workspace layout (floats): coords[B*N*4] | sq[B*N] | feats[B*N*24]  (~3.6 MB)
    float* ws     = (float*)d_ws;
    float* coords = ws;
    float* sqv    = coords + (size_t)B_ * N_ * 4;
    float* feats  = sqv    + (size_t)B_ * N_;

    gravnet_proj<<<dim3((B_ * N_ + 255) / 256), dim3(256), 0, stream>>>(
        x, Wf, bf, Ws, bs, coords, sqv, feats);

    gravnet_main<<<dim3(N_ / 64, B_), dim3(64), 0, stream>>>(
        x, Wo, bo, coords, sqv, feats, outp);
}